// Aggregate_71725953843784
// MI455X (gfx1250) — compile-verified
//
#include <hip/hip_runtime.h>

// GraphSAGE 2-layer mean aggregator for MI455X (gfx1250, wave32).
// fp32 end-to-end using V_WMMA_F32_16X16X4_F32 (problem is gather-BW bound;
// fp32 WMMA preserves reference numerics at zero conversion cost).

typedef __attribute__((ext_vector_type(2))) float v2f;
typedef __attribute__((ext_vector_type(8))) float v8f;

#define D_FEAT 128
#define OUT0   128
#define BATCH  1024
#define S1     25
#define S2     10
#define H      256      // concat width = 2*OUT0 = 2*OUT1
#define LDX0   132      // padded LDS stride for K=128 tiles (bank-conflict-free)
#define LDX1   260      // padded LDS stride for K=256 tiles

// 16x16 f32 output tile: acc += X(16xK from LDS) * W(K x ldw, global) columns [colBase, colBase+16)
template <int K, int LDX>
__device__ __forceinline__ v8f wmma_tile_f32(const float* Xlds,
                                             const float* __restrict__ W,
                                             int ldw, int colBase, int lane) {
  const int n  = colBase + (lane & 15);
  const int m  = lane & 15;
  const int kh = (lane >> 4) << 1;  // lanes 0-15 -> K pair {0,1}; lanes 16-31 -> {2,3}
  v8f acc = {0.f, 0.f, 0.f, 0.f, 0.f, 0.f, 0.f, 0.f};
#pragma unroll
  for (int k0 = 0; k0 < K; k0 += 4) {
    const int ka = k0 + kh;
    v2f a, b;
    a.x = Xlds[m * LDX + ka];
    a.y = Xlds[m * LDX + ka + 1];
    b.x = W[(size_t)ka * ldw + n];
    b.y = W[(size_t)(ka + 1) * ldw + n];
    // (neg_a, A, neg_b, B, c_mod, C, reuse_a, reuse_b)
    acc = __builtin_amdgcn_wmma_f32_16x16x4_f32(false, a, false, b, (short)0,
                                                acc, false, false);
  }
  return acc;
}

__device__ __forceinline__ void store_relu(float* dst, int ldd, int col, v8f acc,
                                           int lane) {
  const int n   = lane & 15;
  const int mhi = (lane >> 4) * 8;   // C/D: VGPR r holds M=r (lanes 0-15) / M=8+r (lanes 16-31)
#pragma unroll
  for (int r = 0; r < 8; ++r) {
    float v = acc[r];
    dst[(size_t)(mhi + r) * ldd + col + n] = fmaxf(v, 0.0f);
  }
}

// ---- layer 0, inner hop: n1[r] = relu(concat(h1[r]@Ws0, mean_j h2[r,j]@Wn0)), r in [0,25600)
__global__ __launch_bounds__(256) void sage_n1(const float* __restrict__ features,
                                               const float* __restrict__ w_self0,
                                               const float* __restrict__ w_neigh0,
                                               const int* __restrict__ samples1,
                                               const int* __restrict__ samples2,
                                               float* __restrict__ n1) {
  __shared__ float Xs[16 * LDX0];
  __shared__ float Xn[16 * LDX0];
  const int tid     = threadIdx.x;
  const int rowBase = blockIdx.x * 16;

  {  // gather phase: 16 threads/row, 8 floats each
    const int row = tid >> 4;
    const int c0  = (tid & 15) * 8;
    const int r   = rowBase + row;
    const size_t si = (size_t)samples1[r];
    const float4* fs = (const float4*)(features + si * D_FEAT + c0);
    float4 sa = fs[0], sb = fs[1];
    *(float4*)&Xs[row * LDX0 + c0]     = sa;
    *(float4*)&Xs[row * LDX0 + c0 + 4] = sb;
    float accv[8] = {0.f, 0.f, 0.f, 0.f, 0.f, 0.f, 0.f, 0.f};
    for (int j = 0; j < S2; ++j) {
      const size_t ni = (size_t)samples2[r * S2 + j];
      const float4* fn = (const float4*)(features + ni * D_FEAT + c0);
      float4 x = fn[0], y = fn[1];
      accv[0] += x.x; accv[1] += x.y; accv[2] += x.z; accv[3] += x.w;
      accv[4] += y.x; accv[5] += y.y; accv[6] += y.z; accv[7] += y.w;
    }
    const float inv = 1.0f / (float)S2;
#pragma unroll
    for (int t = 0; t < 8; ++t) Xn[row * LDX0 + c0 + t] = accv[t] * inv;
  }
  __syncthreads();

  const int wave = tid >> 5, lane = tid & 31;
  const int colBase = wave * 16;
  v8f accS = wmma_tile_f32<D_FEAT, LDX0>(Xs, w_self0, OUT0, colBase, lane);
  v8f accN = wmma_tile_f32<D_FEAT, LDX0>(Xn, w_neigh0, OUT0, colBase, lane);
  float* dst = n1 + (size_t)rowBase * H;
  store_relu(dst, H, colBase, accS, lane);
  store_relu(dst, H, OUT0 + colBase, accN, lane);
}

// ---- layer 0, root hop: n0[b] = relu(concat(h0[b]@Ws0, mean_s h1[b,s]@Wn0)), b in [0,1024)
__global__ __launch_bounds__(256) void sage_n0(const float* __restrict__ features,
                                               const float* __restrict__ w_self0,
                                               const float* __restrict__ w_neigh0,
                                               const int* __restrict__ samples0,
                                               const int* __restrict__ samples1,
                                               float* __restrict__ n0) {
  __shared__ float Xs[16 * LDX0];
  __shared__ float Xn[16 * LDX0];
  const int tid     = threadIdx.x;
  const int rowBase = blockIdx.x * 16;

  {
    const int row = tid >> 4;
    const int c0  = (tid & 15) * 8;
    const int r   = rowBase + row;
    const size_t si = (size_t)samples0[r];
    const float4* fs = (const float4*)(features + si * D_FEAT + c0);
    float4 sa = fs[0], sb = fs[1];
    *(float4*)&Xs[row * LDX0 + c0]     = sa;
    *(float4*)&Xs[row * LDX0 + c0 + 4] = sb;
    float accv[8] = {0.f, 0.f, 0.f, 0.f, 0.f, 0.f, 0.f, 0.f};
    for (int j = 0; j < S1; ++j) {
      const size_t ni = (size_t)samples1[r * S1 + j];
      const float4* fn = (const float4*)(features + ni * D_FEAT + c0);
      float4 x = fn[0], y = fn[1];
      accv[0] += x.x; accv[1] += x.y; accv[2] += x.z; accv[3] += x.w;
      accv[4] += y.x; accv[5] += y.y; accv[6] += y.z; accv[7] += y.w;
    }
    const float inv = 1.0f / (float)S1;
#pragma unroll
    for (int t = 0; t < 8; ++t) Xn[row * LDX0 + c0 + t] = accv[t] * inv;
  }
  __syncthreads();

  const int wave = tid >> 5, lane = tid & 31;
  const int colBase = wave * 16;
  v8f accS = wmma_tile_f32<D_FEAT, LDX0>(Xs, w_self0, OUT0, colBase, lane);
  v8f accN = wmma_tile_f32<D_FEAT, LDX0>(Xn, w_neigh0, OUT0, colBase, lane);
  float* dst = n0 + (size_t)rowBase * H;
  store_relu(dst, H, colBase, accS, lane);
  store_relu(dst, H, OUT0 + colBase, accN, lane);
}

// ---- layer 1: out[b] = relu(concat(n0[b]@Ws1, mean_s n1[b,s]@Wn1)), K=256
__global__ __launch_bounds__(256) void sage_out(const float* __restrict__ n0,
                                                const float* __restrict__ n1,
                                                const float* __restrict__ w_self1,
                                                const float* __restrict__ w_neigh1,
                                                float* __restrict__ out) {
  __shared__ float Xs[16 * LDX1];
  __shared__ float Xn[16 * LDX1];
  const int tid     = threadIdx.x;
  const int rowBase = blockIdx.x * 16;

  {  // gather phase: 16 threads/row, 16 floats each (K=256)
    const int row = tid >> 4;
    const int c0  = (tid & 15) * 16;
    const int r   = rowBase + row;
    const float4* fs = (const float4*)(n0 + (size_t)r * H + c0);
#pragma unroll
    for (int q = 0; q < 4; ++q) *(float4*)&Xs[row * LDX1 + c0 + 4 * q] = fs[q];
    float accv[16];
#pragma unroll
    for (int t = 0; t < 16; ++t) accv[t] = 0.f;
    for (int s = 0; s < S1; ++s) {
      const float4* fn = (const float4*)(n1 + (size_t)(r * S1 + s) * H + c0);
#pragma unroll
      for (int q = 0; q < 4; ++q) {
        float4 x = fn[q];
        accv[4 * q + 0] += x.x; accv[4 * q + 1] += x.y;
        accv[4 * q + 2] += x.z; accv[4 * q + 3] += x.w;
      }
    }
    const float inv = 1.0f / (float)S1;
#pragma unroll
    for (int t = 0; t < 16; ++t) Xn[row * LDX1 + c0 + t] = accv[t] * inv;
  }
  __syncthreads();

  const int wave = tid >> 5, lane = tid & 31;
  const int colBase = wave * 16;
  v8f accS = wmma_tile_f32<H, LDX1>(Xs, w_self1, OUT0, colBase, lane);
  v8f accN = wmma_tile_f32<H, LDX1>(Xn, w_neigh1, OUT0, colBase, lane);
  float* dst = out + (size_t)rowBase * H;
  store_relu(dst, H, colBase, accS, lane);
  store_relu(dst, H, OUT0 + colBase, accN, lane);
}

extern "C" void kernel_launch(void* const* d_in, const int* in_sizes, int n_in,
                              void* d_out, int out_size, void* d_ws, size_t ws_size,
                              hipStream_t stream) {
  (void)in_sizes; (void)n_in; (void)out_size; (void)ws_size;
  const float* features = (const float*)d_in[0];
  const float* w_self0  = (const float*)d_in[1];
  const float* w_neigh0 = (const float*)d_in[2];
  const float* w_self1  = (const float*)d_in[3];
  const float* w_neigh1 = (const float*)d_in[4];
  const int* samples0   = (const int*)d_in[5];
  const int* samples1   = (const int*)d_in[6];
  const int* samples2   = (const int*)d_in[7];
  float* out = (float*)d_out;

  // workspace layout: n1 [25600 x 256] then n0 [1024 x 256]  (27.3 MB total)
  float* n1 = (float*)d_ws;
  float* n0 = n1 + (size_t)BATCH * S1 * H;

  sage_n1<<<dim3((BATCH * S1) / 16), dim3(256), 0, stream>>>(
      features, w_self0, w_neigh0, samples1, samples2, n1);
  sage_n0<<<dim3(BATCH / 16), dim3(256), 0, stream>>>(
      features, w_self0, w_neigh0, samples0, samples1, n0);
  sage_out<<<dim3(BATCH / 16), dim3(256), 0, stream>>>(
      n0, n1, w_self1, w_neigh1, out);
}